// DozerAnomalyAttention_89781996356420
// MI455X (gfx1250) — compile-verified
//
#include <hip/hip_runtime.h>
#include <math.h>

typedef __attribute__((ext_vector_type(2))) float v2f;
typedef __attribute__((ext_vector_type(8))) float v8f;

#define B_ 4
#define L_ 1024
#define H_ 8
#define E_ 64
#define LOCAL_WINDOW_ 64
#define STRIDE_ 16

__device__ __forceinline__ bool keep_mask(int i, int j) {
    if (j > i) return false;                    // causal
    int dist = i - j;
    return (dist <= LOCAL_WINDOW_ / 2) || (dist % (STRIDE_ + 1) == 0);
}

// One workgroup (4 waves / 128 threads) computes a 16-row strip of series
// and V for one (b,h).
__global__ __launch_bounds__(128)
void dozer_attn_kernel(const float* __restrict__ Q,
                       const float* __restrict__ K,
                       const float* __restrict__ Vv,
                       float* __restrict__ outV,
                       float* __restrict__ outS) {
    const int it   = blockIdx.x;        // row-tile index (L/16)
    const int h    = blockIdx.y;
    const int b    = blockIdx.z;
    const int i0   = it * 16;
    const int tid  = threadIdx.x;
    const int wave = tid >> 5;
    const int lane = tid & 31;
    const int half = lane >> 4;         // 0: lanes 0-15, 1: lanes 16-31
    const int lm   = lane & 15;

    __shared__ float sQ[16][72];        // padded: avoid bank conflicts
    __shared__ float sS[16][L_ + 1];    // 16 x 1024 score/prob strip (padded)

    // ---- load Q tile (16 x 64) into LDS: 128 threads x 8 elements ----
    {
        int r  = tid >> 3;
        int e0 = (tid & 7) * 8;
        const float* qp = Q + (((size_t)b * L_ + (i0 + r)) * H_ + h) * E_ + e0;
#pragma unroll
        for (int e = 0; e < 8; ++e) sQ[r][e0 + e] = qp[e];
    }
    __syncthreads();

    const float scale   = 0.125f;       // 1/sqrt(64)
    const float NEG_INF = -__builtin_inff();

    // ---- scores: wave w owns column tiles jt = w, w+4, ... ----
    for (int jt = wave; jt < L_ / 16; jt += 4) {
        int j0 = jt * 16;
        if (j0 > i0 + 15) {             // fully in the causal-masked future
#pragma unroll
            for (int r = 0; r < 8; ++r)
                sS[r + 8 * half][j0 + lm] = NEG_INF;
            continue;
        }
        v8f c = {0.f, 0.f, 0.f, 0.f, 0.f, 0.f, 0.f, 0.f};
        const float* kp = K + (((size_t)b * L_ + (j0 + lm)) * H_ + h) * E_;
#pragma unroll
        for (int kk = 0; kk < E_; kk += 4) {
            int e0 = kk + 2 * half;
            v2f a, bb;
            a.x  = sQ[lm][e0];
            a.y  = sQ[lm][e0 + 1];
            bb.x = kp[e0];              // B[k][n] = Key[j0+n][e0+k]
            bb.y = kp[e0 + 1];
            c = __builtin_amdgcn_wmma_f32_16x16x4_f32(
                    false, a, false, bb, (short)0, c, false, false);
        }
#pragma unroll
        for (int r = 0; r < 8; ++r) {
            int i = i0 + r + 8 * half;
            int j = j0 + lm;
            sS[r + 8 * half][j0 + lm] =
                keep_mask(i, j) ? (scale * c[r]) : NEG_INF;
        }
    }
    __syncthreads();

    // ---- row softmax: wave w handles rows 4w..4w+3, lane-strided cols ----
    for (int rr = 0; rr < 4; ++rr) {
        int r = wave * 4 + rr;
        float m = NEG_INF;
        for (int cidx = lane; cidx < L_; cidx += 32)
            m = fmaxf(m, sS[r][cidx]);
#pragma unroll
        for (int o = 16; o > 0; o >>= 1)
            m = fmaxf(m, __shfl_xor(m, o, 32));
        float sum = 0.f;
        for (int cidx = lane; cidx < L_; cidx += 32) {
            float e = __expf(sS[r][cidx] - m);   // exp(-inf)=0 handles mask
            sS[r][cidx] = e;
            sum += e;
        }
#pragma unroll
        for (int o = 16; o > 0; o >>= 1)
            sum += __shfl_xor(sum, o, 32);
        float inv = 1.0f / sum;
        float* srow = outS + (((size_t)b * H_ + h) * L_ + (i0 + r)) * (size_t)L_;
        for (int cidx = lane; cidx < L_; cidx += 32) {
            float p = sS[r][cidx] * inv;
            sS[r][cidx] = p;
            srow[cidx] = p;                       // coalesced series write
        }
    }
    __syncthreads();

    // ---- V = P (16x1024) x values (1024x64): wave w owns e-tile w*16 ----
    {
        const int e0 = wave * 16;
        v8f c = {0.f, 0.f, 0.f, 0.f, 0.f, 0.f, 0.f, 0.f};
        const int kmax = i0 + 16;       // P is exactly zero past the causal edge
        const size_t vstride = (size_t)H_ * E_;   // floats between key rows
        const float* vbase = Vv + (((size_t)b * L_) * H_ + h) * E_ + e0 + lm;
        for (int kk = 0; kk < kmax; kk += 4) {
            int k0 = kk + 2 * half;
            v2f a, bb;
            a.x  = sS[lm][k0];
            a.y  = sS[lm][k0 + 1];
            bb.x = vbase[(size_t)k0 * vstride];
            bb.y = vbase[(size_t)(k0 + 1) * vstride];
            c = __builtin_amdgcn_wmma_f32_16x16x4_f32(
                    false, a, false, bb, (short)0, c, false, false);
        }
#pragma unroll
        for (int r = 0; r < 8; ++r) {
            int l = i0 + r + 8 * half;
            outV[(((size_t)b * L_ + l) * H_ + h) * E_ + e0 + lm] = c[r];
        }
    }
}

// One block per (b,h,i) row: compute sig once, stream prior and sigma_out.
__global__ __launch_bounds__(256)
void dozer_prior_kernel(const float* __restrict__ sigma,
                        float* __restrict__ outPrior,
                        float* __restrict__ outSigma) {
    const int i = blockIdx.x;
    const int h = blockIdx.y;
    const int b = blockIdx.z;

    float x  = sigma[((size_t)b * L_ + i) * H_ + h];
    float sg = 1.0f / (1.0f + __expf(-5.0f * x)) + 1e-5f;   // sigmoid(5x)+1e-5
    sg = __expf(sg * 1.0986122886681098f) - 1.0f;           // 3^sg - 1
    float cl     = fmaxf(sg, 1e-8f);
    float inv2c2 = 1.0f / (2.0f * cl * cl);
    float coef   = 0.3989422804014327f / cl;                // 1/(sqrt(2pi)*cl)

    size_t rowoff = (((size_t)b * H_ + h) * L_ + i) * (size_t)L_;
    for (int j = threadIdx.x; j < L_; j += blockDim.x) {
        float d = (float)(i - j);
        outPrior[rowoff + j] = __expf(-d * d * inv2c2) * coef;
        outSigma[rowoff + j] = sg;
    }
}

extern "C" void kernel_launch(void* const* d_in, const int* in_sizes, int n_in,
                              void* d_out, int out_size, void* d_ws, size_t ws_size,
                              hipStream_t stream) {
    (void)in_sizes; (void)n_in; (void)out_size; (void)d_ws; (void)ws_size;

    const float* Q     = (const float*)d_in[0];
    const float* K     = (const float*)d_in[1];
    const float* Vv    = (const float*)d_in[2];
    const float* sigma = (const float*)d_in[3];

    float* out    = (float*)d_out;
    float* outV   = out;                                   // [B,L,H,E]
    float* outS   = outV + (size_t)B_ * L_ * H_ * E_;      // [B,H,L,L]
    float* outP   = outS + (size_t)B_ * H_ * L_ * L_;      // [B,H,L,L]
    float* outSig = outP + (size_t)B_ * H_ * L_ * L_;      // [B,H,L,L]

    dim3 g1(L_ / 16, H_, B_);
    dozer_attn_kernel<<<g1, 128, 0, stream>>>(Q, K, Vv, outV, outS);

    dim3 g2(L_, H_, B_);
    dozer_prior_kernel<<<g2, 256, 0, stream>>>(sigma, outP, outSig);
}